// HGT_HLS_87084756893853
// MI455X (gfx1250) — compile-verified
//
#include <hip/hip_runtime.h>
#include <math.h>

// ---------------------------------------------------------------------------
// CDNA5 (gfx1250) implementation of the HGT + JK-LSTM + SAG-pool reference.
// Dense math: V_WMMA_F32_16X16X4_F32 (full fp32 matrix pipe).
// A-tiles for dense GEMMs are staged into LDS by the Tensor Data Mover
// (tensor_load_to_lds, double-buffered, LDS padding via descriptor), gathered
// edge GEMMs use cooperative b128 loads.
// ---------------------------------------------------------------------------

typedef float v2f __attribute__((ext_vector_type(2)));
typedef float v8f __attribute__((ext_vector_type(8)));
typedef unsigned u32x4 __attribute__((ext_vector_type(4)));
typedef int i32x4 __attribute__((ext_vector_type(4)));
typedef int i32x8 __attribute__((ext_vector_type(8)));

#define WAVES 8     // waves per block (wave32 -> 256 threads)
#define APITCH 20   // LDS row pitch (20 dwords: 16 data + 4 pad, 16B aligned)

__device__ __forceinline__ float gelu_t(float x) {
  float x3 = x * x * x;
  return 0.5f * x * (1.0f + tanhf(0.7978845608028654f * (x + 0.044715f * x3)));
}
__device__ __forceinline__ float sigm(float x) { return 1.0f / (1.0f + expf(-x)); }

// order-preserving float<->uint encode for atomicMax on floats
__device__ __forceinline__ unsigned encf(float f) {
  unsigned u = __float_as_uint(f);
  return (u & 0x80000000u) ? ~u : (u | 0x80000000u);
}
__device__ __forceinline__ float decf(unsigned u) {
  u = (u & 0x80000000u) ? (u & 0x7fffffffu) : ~u;
  return __uint_as_float(u);
}

// ---------------------------------------------------------------------------
// TDM: DMA a 128-row x 16-col fp32 tile (row stride lda) into LDS with a
// 4-dword pad after every 16 dwords (=> LDS pitch 20). Rows beyond
// rowsRemain are zero-filled by tensor OOB handling.
// D# layout per CDNA5 ISA 8.3/8.4.
// ---------------------------------------------------------------------------
__device__ __forceinline__ void tdm_load_tile(const float* gsrc, unsigned ldsOff,
                                              int lda, int rowsRemain) {
  unsigned long long ga = (unsigned long long)(const void*)gsrc;
  u32x4 g0;
  g0[0] = 1u;                                                   // count=1
  g0[1] = ldsOff;                                               // lds byte addr
  g0[2] = (unsigned)ga;                                         // global addr lo
  g0[3] = (unsigned)((ga >> 32) & 0x01FFFFFFull) | (2u << 30);  // addr hi | type=2
  i32x8 g1;
  g1[0] = (int)((2u << 16) | (1u << 20) | (3u << 22) | (3u << 25));
  //           data_size=4B  pad_enable   every 16 dw   pad 4 dw
  g1[1] = (int)(16u << 16);                                // tensor_dim0 = 16
  unsigned td1 = (unsigned)rowsRemain;                     // tensor_dim1
  g1[2] = (int)((td1 & 0xFFFFu) << 16);
  g1[3] = (int)((td1 >> 16) | (16u << 16));                // | tile_dim0 = 16
  g1[4] = 128;                                             // tile_dim1=128, tile_dim2=0
  g1[5] = lda;                                             // tensor_dim0_stride
  g1[6] = 0;
  g1[7] = 0;
  i32x4 z4 = {0, 0, 0, 0};
#if __clang_major__ >= 23
  i32x8 z8 = {0, 0, 0, 0, 0, 0, 0, 0};
  __builtin_amdgcn_tensor_load_to_lds(g0, g1, z4, z4, z8, 0);
#else
  __builtin_amdgcn_tensor_load_to_lds(g0, g1, z4, z4, 0);
#endif
}

// ---------------------------------------------------------------------------
// WMMA GEMM:  C[M,N] = (A[M,K] or A[rowIdx[m],K]) @ B[K,N]  (+bias | +=C)
// Each wave owns a 16 x (16*NT) output tile. A is staged 128x16 in LDS:
//   USE_TDM=1 : tensor_load_to_lds, double-buffered (DMA overlaps WMMA)
//   USE_TDM=0 : cooperative b128 loads with optional row gather
// B fragments for the whole K-tile are pre-loaded into registers (one load
// clause -> 4*NT back-to-back WMMAs). K%16==0, N%(16*NT)==0 by construction.
// ---------------------------------------------------------------------------
template <int NT, int USE_TDM>
__global__ __launch_bounds__(256) void wmma_gemm(
    const float* __restrict__ A, const int* __restrict__ rowIdx, int lda,
    const float* __restrict__ B, int ldb, float* __restrict__ C, int ldc,
    const float* __restrict__ bias, int M, int K, int accumulate) {
  __shared__ __align__(16) float As[2][WAVES * 16][APITCH];
  const int tid = threadIdx.x;
  const int wave = tid >> 5, lane = tid & 31;
  const int half = lane >> 4, lr = lane & 15;
  const int mtile = blockIdx.x * WAVES + wave;
  const int colbase = blockIdx.y * (16 * NT);
  const long mrow0 = (long)mtile * 16;
  const bool mvalid = (mrow0 < (long)M);
  const long mbase = mvalid ? mrow0 : 0;  // clamp so all waves run WMMA (EXEC all-1s)
  const long rowbase = (long)blockIdx.x * (WAVES * 16);

  v8f acc[NT];
  if (accumulate) {
#pragma unroll
    for (int t = 0; t < NT; ++t)
#pragma unroll
      for (int r = 0; r < 8; ++r)
        acc[t][r] = C[(mbase + half * 8 + r) * (long)ldc + colbase + t * 16 + lr];
  } else if (bias) {
#pragma unroll
    for (int t = 0; t < NT; ++t) {
      float bv = bias[colbase + t * 16 + lr];
#pragma unroll
      for (int r = 0; r < 8; ++r) acc[t][r] = bv;
    }
  } else {
#pragma unroll
    for (int t = 0; t < NT; ++t)
#pragma unroll
      for (int r = 0; r < 8; ++r) acc[t][r] = 0.0f;
  }

  int ibuf = 0;
  if constexpr (USE_TDM) {
    if (wave == 0)
      tdm_load_tile(A + rowbase * lda, (unsigned)(unsigned long long)&As[0][0][0], lda,
                    (int)((long)M - rowbase));
  }

  for (int k0 = 0; k0 < K; k0 += 16) {
    if constexpr (USE_TDM) {
      if (wave == 0) __builtin_amdgcn_s_wait_tensorcnt(0);  // tile k0 landed
      __syncthreads();
      if (wave == 0 && k0 + 16 < K)  // start DMA of next tile into other buffer
        tdm_load_tile(A + rowbase * lda + k0 + 16,
                      (unsigned)(unsigned long long)&As[ibuf ^ 1][0][0], lda,
                      (int)((long)M - rowbase));
    } else {
      __syncthreads();  // protect As reuse
      for (int i = tid; i < WAVES * 16 * 4; i += 256) {  // 512 float4
        int r = i >> 2, c4 = (i & 3) * 4;
        long grow = rowbase + r;
        float4 v = make_float4(0.0f, 0.0f, 0.0f, 0.0f);
        if (grow < (long)M) {
          long ar = rowIdx ? (long)rowIdx[grow] : grow;
          const float* ap = A + ar * (long)lda + k0 + c4;
          v = *(const float4*)ap;
          if (k0 + 16 < K) __builtin_prefetch(ap + 16, 0, 3);  // global_prefetch_b8
        }
        *(float4*)&As[0][r][c4] = v;
      }
      __syncthreads();
    }

    // pre-load all B fragments of this K-tile (one load clause)
    v2f bf[4][NT];
#pragma unroll
    for (int kk = 0; kk < 4; ++kk) {
      const float* bp = B + (long)(k0 + kk * 4 + 2 * half) * ldb + colbase + lr;
#pragma unroll
      for (int t = 0; t < NT; ++t) {
        bf[kk][t][0] = bp[t * 16];
        bf[kk][t][1] = bp[t * 16 + ldb];
      }
    }

    float(*Ac)[APITCH] = As[USE_TDM ? ibuf : 0];
#pragma unroll
    for (int kk = 0; kk < 4; ++kk) {
      v2f a;
      a[0] = Ac[wave * 16 + lr][kk * 4 + 2 * half];
      a[1] = Ac[wave * 16 + lr][kk * 4 + 2 * half + 1];
#pragma unroll
      for (int t = 0; t < NT; ++t)
        acc[t] = __builtin_amdgcn_wmma_f32_16x16x4_f32(false, a, false, bf[kk][t],
                                                       (short)0, acc[t], false, false);
    }
    if constexpr (USE_TDM) ibuf ^= 1;
  }

  if (mvalid) {
#pragma unroll
    for (int t = 0; t < NT; ++t)
#pragma unroll
      for (int r = 0; r < 8; ++r)
        C[(mrow0 + half * 8 + r) * (long)ldc + colbase + t * 16 + lr] = acc[t][r];
  }
}

// ---------------------------------------------------------------------------
// elementwise / fill helpers
// ---------------------------------------------------------------------------
__global__ void fill_f32(float* p, float v, int n) {
  int i = blockIdx.x * blockDim.x + threadIdx.x;
  if (i < n) p[i] = v;
}
__global__ void fill_u32(unsigned* p, unsigned v, int n) {
  int i = blockIdx.x * blockDim.x + threadIdx.x;
  if (i < n) p[i] = v;
}
__global__ void elem_gelu(float* p, int n) {
  int i = blockIdx.x * blockDim.x + threadIdx.x;
  if (i < n) p[i] = gelu_t(p[i]);
}
__global__ void skip_mix(float* o, const float* __restrict__ xprev,
                         const float* __restrict__ skip, int n) {
  int i = blockIdx.x * blockDim.x + threadIdx.x;
  if (i >= n) return;
  float beta = sigm(skip[0]);
  o[i] = beta * o[i] + (1.0f - beta) * xprev[i];
}

// ---------------------------------------------------------------------------
// edge softmax passes (segment softmax over incoming edges per dst node)
// kh: per-head relation-transformed keys/values, [E,32]
// ---------------------------------------------------------------------------
__global__ void edge_score(const float* __restrict__ kh, const float* __restrict__ Q,
                           const int* __restrict__ dst, const float* __restrict__ prel,
                           int h, float* __restrict__ score, unsigned* __restrict__ smaxenc,
                           int E) {
  int e = blockIdx.x * blockDim.x + threadIdx.x;
  if (e >= E) return;
  int d = dst[e];
  const float* kp = kh + (long)e * 32;
  const float* qp = Q + (long)d * 128 + h * 32;
  float s = 0.0f;
  for (int j = 0; j < 32; ++j) s += kp[j] * qp[j];
  s *= prel[h] * 0.17677669529663687f;  // 1/sqrt(32)
  score[(long)e * 4 + h] = s;
  atomicMax(&smaxenc[(long)d * 4 + h], encf(s));
}

__global__ void smax_fix(unsigned* p, int n) {  // decode in place; -inf -> 0
  int i = blockIdx.x * blockDim.x + threadIdx.x;
  if (i >= n) return;
  float f = decf(p[i]);
  if (!isfinite(f)) f = 0.0f;
  ((float*)p)[i] = f;
}

__global__ void edge_exp(float* __restrict__ score, const float* __restrict__ smax,
                         const int* __restrict__ dst, float* __restrict__ den, int E) {
  int idx = blockIdx.x * blockDim.x + threadIdx.x;
  if (idx >= E * 4) return;
  int e = idx >> 2, h = idx & 3;
  int d = dst[e];
  float ex = expf(score[idx] - smax[(long)d * 4 + h]);
  score[idx] = ex;
  atomicAdd(&den[(long)d * 4 + h], ex);
}

__global__ void edge_aggr(const float* __restrict__ mh, const float* __restrict__ ex,
                          const float* __restrict__ den, const int* __restrict__ dst,
                          int h, float* __restrict__ aggr, int E) {
  int e = blockIdx.x * blockDim.x + threadIdx.x;
  if (e >= E) return;
  int d = dst[e];
  float a = ex[(long)e * 4 + h] / (den[(long)d * 4 + h] + 1e-16f);
  const float* mp = mh + (long)e * 32;
  float* ap = aggr + (long)d * 128 + h * 32;
  for (int j = 0; j < 32; ++j) atomicAdd(&ap[j], a * mp[j]);
}

// ---------------------------------------------------------------------------
// LayerNorm + gelu, in place, one node per 128-thread block
// ---------------------------------------------------------------------------
__global__ void ln_gelu(float* __restrict__ x, const float* __restrict__ g,
                        const float* __restrict__ b, long rowoff) {
  long row = rowoff + blockIdx.x;
  int j = threadIdx.x;
  float v = x[row * 128 + j];
  float s = v, s2 = v * v;
  for (int m = 16; m >= 1; m >>= 1) {
    s += __shfl_xor(s, m, 32);
    s2 += __shfl_xor(s2, m, 32);
  }
  __shared__ float rs[4], rs2[4];
  int wid = j >> 5;
  if ((j & 31) == 0) { rs[wid] = s; rs2[wid] = s2; }
  __syncthreads();
  float ts = rs[0] + rs[1] + rs[2] + rs[3];
  float t2 = rs2[0] + rs2[1] + rs2[2] + rs2[3];
  float mean = ts * (1.0f / 128.0f);
  float var = t2 * (1.0f / 128.0f) - mean * mean;
  float y = (v - mean) * rsqrtf(var + 1e-5f) * g[j] + b[j];
  x[row * 128 + j] = gelu_t(y);
}

// ---------------------------------------------------------------------------
// JK LSTM pieces
// ---------------------------------------------------------------------------
__global__ void lstm_gate(const float* __restrict__ G, float* __restrict__ h,
                          float* __restrict__ c, int n) {
  int idx = blockIdx.x * blockDim.x + threadIdx.x;
  if (idx >= n * 256) return;
  int node = idx >> 8, j = idx & 255;
  const float* g = G + (long)node * 1024;
  float ii = g[j], ff = g[256 + j], gg = g[512 + j], oo = g[768 + j];
  float cc = sigm(ff) * c[idx] + sigm(ii) * tanhf(gg);
  c[idx] = cc;
  h[idx] = sigm(oo) * tanhf(cc);
}

__global__ void att_dot(const float* __restrict__ h, const float* __restrict__ Wa,
                        float* __restrict__ att, int n, int add) {
  int w = (blockIdx.x * blockDim.x + threadIdx.x) >> 5;
  int lane = threadIdx.x & 31;
  if (w >= n) return;
  float s = 0.0f;
  for (int k = lane; k < 256; k += 32) s += h[(long)w * 256 + k] * Wa[k];
  for (int m = 16; m >= 1; m >>= 1) s += __shfl_xor(s, m, 32);
  if (lane == 0) {
    if (add) att[w] += s;
    else att[w] = s;
  }
}

__global__ void jk_finish(const float* __restrict__ att, int CHs,
                          const float* __restrict__ ba, const float* __restrict__ o0,
                          const float* __restrict__ o1, const float* __restrict__ o2,
                          const float* __restrict__ o3, long rowoff,
                          float* __restrict__ xjk) {
  int node = blockIdx.x;
  int j = threadIdx.x;  // 128
  float bb = ba[0];
  float a0 = att[0 * CHs + node] + bb, a1 = att[1 * CHs + node] + bb;
  float a2 = att[2 * CHs + node] + bb, a3 = att[3 * CHs + node] + bb;
  float mx = fmaxf(fmaxf(a0, a1), fmaxf(a2, a3));
  float e0 = expf(a0 - mx), e1 = expf(a1 - mx), e2 = expf(a2 - mx), e3 = expf(a3 - mx);
  float inv = 1.0f / (e0 + e1 + e2 + e3);
  long r = (rowoff + node) * 128 + j;
  xjk[r] = (e0 * o0[r] + e1 * o1[r] + e2 * o2[r] + e3 * o3[r]) * inv;
}

// ---------------------------------------------------------------------------
// SAG pooling + MLP
// ---------------------------------------------------------------------------
__global__ void pool_score(const float* __restrict__ xjk, const float* __restrict__ wi,
                           const float* __restrict__ bi, const float* __restrict__ wp,
                           const float* __restrict__ bp, int N1, int NN,
                           float* __restrict__ s) {
  int g = blockIdx.x * blockDim.x + threadIdx.x;
  if (g >= NN) return;
  const float* w = (g < N1) ? wi : wp;
  float bb = (g < N1) ? bi[0] : bp[0];
  float acc = 0.0f;
  for (int j = 0; j < 128; ++j) acc += xjk[(long)g * 128 + j] * w[j];
  s[g] = acc + bb;
}

__global__ void top8(const float* __restrict__ s, int n, float* __restrict__ outv,
                     int* __restrict__ outi) {
  const int T = 256;
  float bv[8];
  int bi[8];
  for (int k = 0; k < 8; ++k) { bv[k] = -INFINITY; bi[k] = 0x7fffffff; }
  for (int i = threadIdx.x; i < n; i += T) {
    float v = s[i];
    if (v > bv[7] || (v == bv[7] && i < bi[7])) {
      int k = 7;
      while (k > 0 && (v > bv[k - 1] || (v == bv[k - 1] && i < bi[k - 1]))) {
        bv[k] = bv[k - 1]; bi[k] = bi[k - 1]; --k;
      }
      bv[k] = v; bi[k] = i;
    }
  }
  __shared__ float lv[T * 8];
  __shared__ int li[T * 8];
  for (int k = 0; k < 8; ++k) { lv[threadIdx.x * 8 + k] = bv[k]; li[threadIdx.x * 8 + k] = bi[k]; }
  __syncthreads();
  if (threadIdx.x == 0) {
    for (int p = 0; p < 8; ++p) {
      float best = -INFINITY;
      int bidx = 0x7fffffff, bslot = 0;
      for (int q = 0; q < T * 8; ++q) {
        float v = lv[q];
        if (v > best || (v == best && li[q] < bidx)) { best = v; bidx = li[q]; bslot = q; }
      }
      outv[p] = best;
      outi[p] = bidx;
      lv[bslot] = -INFINITY;
    }
  }
}

__global__ void pool_gather(const float* __restrict__ xjk, const float* __restrict__ vals,
                            const int* __restrict__ idx, float* __restrict__ hvec) {
  int t = blockIdx.x * blockDim.x + threadIdx.x;
  if (t >= 1024) return;
  int p = t >> 7, j = t & 127;
  hvec[t] = xjk[(long)idx[p] * 128 + j] * tanhf(vals[p]);
}

__global__ void mlp_layer(const float* __restrict__ in, const float* __restrict__ W,
                          const float* __restrict__ b, float* __restrict__ out, int IN,
                          int OUT, int act) {
  int j = blockIdx.x * blockDim.x + threadIdx.x;
  if (j >= OUT) return;
  float s = b[j];
  for (int k = 0; k < IN; ++k) s += in[k] * W[(long)k * OUT + j];
  if (act == 1) s = gelu_t(s);
  else if (act == 2) {
    if (isnan(s)) s = 0.0f;
    else if (isinf(s)) s = (s > 0.0f) ? 3.4028234663852886e38f : -3.4028234663852886e38f;
  }
  out[j] = s;
}

// ---------------------------------------------------------------------------
// host side
// ---------------------------------------------------------------------------
static inline int cdiv(long a, long b) { return (int)((a + b - 1) / b); }

static void gemm(hipStream_t st, const float* A, const int* rowIdx, int lda,
                 const float* B, int ldb, float* C, int ldc, const float* bias, int M,
                 int K, int N, int acc) {
  int mb = cdiv(cdiv(M, 16), WAVES);
  if (rowIdx == nullptr) {  // dense: TDM-staged A, 64-wide col tiles
    dim3 g(mb, N / 64);
    wmma_gemm<4, 1><<<g, 256, 0, st>>>(A, rowIdx, lda, B, ldb, C, ldc, bias, M, K, acc);
  } else {  // gathered (edge einsum): N == 32
    dim3 g(mb, N / 32);
    wmma_gemm<2, 0><<<g, 256, 0, st>>>(A, rowIdx, lda, B, ldb, C, ldc, bias, M, K, acc);
  }
}

extern "C" void kernel_launch(void* const* d_in, const int* in_sizes, int n_in,
                              void* d_out, int out_size, void* d_ws, size_t ws_size,
                              hipStream_t stream) {
  (void)n_in; (void)out_size; (void)ws_size;
  const int CIN = 64, C = 128, L = 4, CH = 16384;
  const int N1 = in_sizes[0] / CIN, N2 = in_sizes[1] / CIN, NN = N1 + N2;
  const int EN[3] = {in_sizes[2], in_sizes[4], in_sizes[6]};
  const int* esrc[3] = {(const int*)d_in[2], (const int*)d_in[4], (const int*)d_in[6]};
  const int* edst[3] = {(const int*)d_in[3], (const int*)d_in[5], (const int*)d_in[7]};
  const int s_nt[3] = {0, 0, 1}, d_nt[3] = {0, 1, 0};
  long Emax = EN[0]; if (EN[1] > Emax) Emax = EN[1]; if (EN[2] > Emax) Emax = EN[2];
  const long Etot = (long)EN[0] + EN[1] + EN[2];

  // param leaf indices (pytree: sorted dict keys; lists/tuples in order)
  // 0..7 data; 8..23 jk; 24..131 layers(27 each); 132..139 mlp; 140..151 norm; 152..155 pool
  auto P = [&](int i) { return (const float*)d_in[i]; };

  // ---- workspace carve ----
  char* base = (char*)d_ws;
  size_t off = 0;
  auto alloc = [&](long nfl) -> float* {
    float* p = (float*)(base + off);
    off += (size_t)nfl * 4;
    off = (off + 255) & ~(size_t)255;
    return p;
  };
  float* outs[4];
  for (int l = 0; l < 4; ++l) outs[l] = alloc((long)NN * C);
  float* Kb = alloc((long)NN * C);
  float* Qb = alloc((long)NN * C);
  float* Vb = alloc((long)NN * C);
  float* aggr = alloc((long)NN * C);
  float* kmsg = alloc(Emax * 32);         // per-head relation-transformed K or V
  float* scoreb = alloc(Etot * 4);        // [e0|e1|e2] x H, scores then exp()
  float* smax = alloc((long)NN * 4);      // encoded-uint, then decoded float
  float* den = alloc((long)NN * 4);
  float* xjk = alloc((long)NN * C);
  float* G = alloc((long)CH * 1024);
  float* hb = alloc((long)CH * 256);
  float* cb = alloc((long)CH * 256);
  float* attb = alloc(4 * (long)CH);
  float* sall = alloc(NN);
  float* topv = alloc(8);
  int* topi = (int*)alloc(8);
  float* hvec = alloc(1024);
  float* m1 = alloc(512);
  float* m2 = alloc(256);
  float* m3 = alloc(128);

  long sboff[3] = {0, (long)EN[0] * 4, (long)EN[0] * 4 + (long)EN[1] * 4};

  // ================= HGT conv layers =================
  for (int l = 0; l < L; ++l) {
    const int lb = 24 + l * 27;
    const int cin = (l == 0) ? CIN : C;
    const float* xI = (l == 0) ? P(0) : outs[l - 1];
    const float* xP = (l == 0) ? P(1) : outs[l - 1] + (long)N1 * C;

    // K/Q/V projections (per node type)
    gemm(stream, xI, nullptr, cin, P(lb + 2), C, Kb, C, P(lb + 13), N1, cin, C, 0);
    gemm(stream, xP, nullptr, cin, P(lb + 3), C, Kb + (long)N1 * C, C, P(lb + 14), N2, cin, C, 0);
    gemm(stream, xI, nullptr, cin, P(lb + 4), C, Qb, C, P(lb + 15), N1, cin, C, 0);
    gemm(stream, xP, nullptr, cin, P(lb + 5), C, Qb + (long)N1 * C, C, P(lb + 16), N2, cin, C, 0);
    gemm(stream, xI, nullptr, cin, P(lb + 6), C, Vb, C, P(lb + 17), N1, cin, C, 0);
    gemm(stream, xP, nullptr, cin, P(lb + 7), C, Vb + (long)N1 * C, C, P(lb + 18), N2, cin, C, 0);

    fill_u32<<<cdiv(NN * 4, 256), 256, 0, stream>>>((unsigned*)smax, 0x007FFFFFu, NN * 4);
    fill_f32<<<cdiv(NN * 4, 256), 256, 0, stream>>>(den, 0.0f, NN * 4);
    fill_f32<<<cdiv((long)NN * C, 256), 256, 0, stream>>>(aggr, 0.0f, NN * C);

    // pass 1: scores + segment max
    for (int et = 0; et < 3; ++et) {
      const float* Kbase = Kb + (long)(s_nt[et] ? N1 : 0) * C;
      const float* Qbase = Qb + (long)(d_nt[et] ? N1 : 0) * C;
      unsigned* smb = (unsigned*)smax + (long)(d_nt[et] ? N1 : 0) * 4;
      for (int h = 0; h < 4; ++h) {
        gemm(stream, Kbase + h * 32, esrc[et], C, P(lb + 8 + et) + h * 1024, 32, kmsg, 32,
             nullptr, EN[et], 32, 32, 0);
        edge_score<<<cdiv(EN[et], 256), 256, 0, stream>>>(
            kmsg, Qbase, edst[et], P(lb + 22 + et), h, scoreb + sboff[et], smb, EN[et]);
      }
    }
    smax_fix<<<cdiv(NN * 4, 256), 256, 0, stream>>>((unsigned*)smax, NN * 4);

    // pass 2: exp + segment sum
    for (int et = 0; et < 3; ++et) {
      long doff = (long)(d_nt[et] ? N1 : 0) * 4;
      edge_exp<<<cdiv((long)EN[et] * 4, 256), 256, 0, stream>>>(
          scoreb + sboff[et], smax + doff, edst[et], den + doff, EN[et]);
    }

    // pass 3: messages + weighted scatter-add
    for (int et = 0; et < 3; ++et) {
      const float* Vbase = Vb + (long)(s_nt[et] ? N1 : 0) * C;
      long doff4 = (long)(d_nt[et] ? N1 : 0) * 4;
      float* abase = aggr + (long)(d_nt[et] ? N1 : 0) * C;
      for (int h = 0; h < 4; ++h) {
        gemm(stream, Vbase + h * 32, esrc[et], C, P(lb + 19 + et) + h * 1024, 32, kmsg, 32,
             nullptr, EN[et], 32, 32, 0);
        edge_aggr<<<cdiv(EN[et], 256), 256, 0, stream>>>(
            kmsg, scoreb + sboff[et], den + doff4, edst[et], h, abase, EN[et]);
      }
    }

    // out projection: o = gelu(aggr) @ Wa + ba
    elem_gelu<<<cdiv((long)NN * C, 256), 256, 0, stream>>>(aggr, NN * C);
    gemm(stream, aggr, nullptr, C, P(lb + 0), C, outs[l], C, P(lb + 11), N1, C, C, 0);
    gemm(stream, aggr + (long)N1 * C, nullptr, C, P(lb + 1), C, outs[l] + (long)N1 * C, C,
         P(lb + 12), N2, C, C, 0);

    if (l > 0) {
      skip_mix<<<cdiv((long)N1 * C, 256), 256, 0, stream>>>(outs[l], outs[l - 1],
                                                            P(lb + 25), N1 * C);
      skip_mix<<<cdiv((long)N2 * C, 256), 256, 0, stream>>>(
          outs[l] + (long)N1 * C, outs[l - 1] + (long)N1 * C, P(lb + 26), N2 * C);
    }
    if (l < L - 1) {
      const int nb = 140 + l * 4;
      ln_gelu<<<N1, 128, 0, stream>>>(outs[l], P(nb + 0), P(nb + 1), 0);
      ln_gelu<<<N2, 128, 0, stream>>>(outs[l], P(nb + 2), P(nb + 3), (long)N1);
    }
  }

  // ================= JK bidirectional LSTM + attention =================
  for (int nt = 0; nt < 2; ++nt) {
    const int jb = 8 + nt * 8;  // Wa,Wh_b,Wh_f,Wi_b,Wi_f,b_b,b_f,ba
    const long ntoff = nt ? (long)N1 : 0;
    const int Nt = nt ? N2 : N1;
    for (long c0 = 0; c0 < Nt; c0 += CH) {
      const int cs = (int)(((long)Nt - c0 < CH) ? (Nt - c0) : CH);
      fill_f32<<<cdiv(4 * CH, 256), 256, 0, stream>>>(attb, 0.0f, 4 * CH);
      for (int dir = 0; dir < 2; ++dir) {
        const float* Wi = P(jb + (dir ? 3 : 4));
        const float* Wh = P(jb + (dir ? 1 : 2));
        const float* bg = P(jb + (dir ? 5 : 6));
        const float* WaJ = P(jb + 0) + (dir ? 256 : 0);
        fill_f32<<<cdiv(cs * 256, 256), 256, 0, stream>>>(hb, 0.0f, cs * 256);
        fill_f32<<<cdiv(cs * 256, 256), 256, 0, stream>>>(cb, 0.0f, cs * 256);
        for (int s = 0; s < 4; ++s) {
          const int t = dir ? 3 - s : s;
          gemm(stream, outs[t] + (ntoff + c0) * C, nullptr, C, Wi, 1024, G, 1024, bg, cs,
               C, 1024, 0);
          gemm(stream, hb, nullptr, 256, Wh, 1024, G, 1024, nullptr, cs, 256, 1024, 1);
          lstm_gate<<<cs, 256, 0, stream>>>(G, hb, cb, cs);
          att_dot<<<cdiv((long)cs * 32, 256), 256, 0, stream>>>(hb, WaJ, attb + t * CH,
                                                                cs, dir);
        }
      }
      jk_finish<<<cs, 128, 0, stream>>>(attb, CH, P(jb + 7), outs[0], outs[1], outs[2],
                                        outs[3], ntoff + c0, xjk);
    }
  }

  // ================= SAG pool (global top-8) + MLP =================
  pool_score<<<cdiv(NN, 256), 256, 0, stream>>>(xjk, P(153), P(152), P(155), P(154), N1,
                                                NN, sall);
  top8<<<1, 256, 0, stream>>>(sall, NN, topv, topi);
  pool_gather<<<4, 256, 0, stream>>>(xjk, topv, topi, hvec);
  mlp_layer<<<2, 256, 0, stream>>>(hvec, P(132), P(133), m1, 1024, 512, 1);
  mlp_layer<<<1, 256, 0, stream>>>(m1, P(134), P(135), m2, 512, 256, 1);
  mlp_layer<<<1, 128, 0, stream>>>(m2, P(136), P(137), m3, 256, 128, 1);
  mlp_layer<<<1, 32, 0, stream>>>(m3, P(138), P(139), (float*)d_out, 128, 1, 2);
}